// Per_Region_Normalization_33775622816279
// MI455X (gfx1250) — compile-verified
//
#include <hip/hip_runtime.h>
#include <hip/hip_bf16.h>

typedef __attribute__((ext_vector_type(2))) float v2f;
typedef __attribute__((ext_vector_type(8))) float v8f;

#define Bn 4
#define Cc 256
#define Ss 256
#define Hh 128
#define Ww 128
#define Jj 8
#define HW (Hh * Ww)          // 16384
#define BHW (Bn * HW)         // 65536
#define NN (Bn * Jj)          // 32  (GEMM N dim)
#define WSTRIDE (Ss * 9)      // 2304 floats per output channel of conv weight

// ---------------- Kernel 1: BN stats -> per-channel scale/shift (b128 loads) ----------------
__global__ void __launch_bounds__(256) k_bn_stats(const float* __restrict__ fp,
                                                  const float* __restrict__ gamma,
                                                  const float* __restrict__ beta,
                                                  float* __restrict__ scale,
                                                  float* __restrict__ shift) {
    int c = blockIdx.x, tid = threadIdx.x;
    __shared__ float ssum[256], ssq[256];
    float s = 0.f, q = 0.f;
    for (int b = 0; b < Bn; ++b) {
        const float4* p = (const float4*)(fp + ((size_t)b * Cc + c) * HW);
        for (int i = tid; i < HW / 4; i += 256) {
            float4 v = p[i];
            s += v.x + v.y + v.z + v.w;
            q += v.x * v.x + v.y * v.y + v.z * v.z + v.w * v.w;
        }
    }
    ssum[tid] = s; ssq[tid] = q;
    __syncthreads();
    for (int off = 128; off > 0; off >>= 1) {
        if (tid < off) { ssum[tid] += ssum[tid + off]; ssq[tid] += ssq[tid + off]; }
        __syncthreads();
    }
    if (tid == 0) {
        const float inv = 1.f / (float)(Bn * HW);
        float mean = ssum[0] * inv;
        float var  = ssq[0] * inv - mean * mean;   // biased, matches reference
        float sc = gamma[c] * rsqrtf(var + 1e-5f);
        scale[c] = sc;
        shift[c] = beta[c] - mean * sc;
    }
}

// ---------------- Kernel 2: one-hot sg -> label map (4 pixels/thread, b128) ----------------
__global__ void __launch_bounds__(256) k_label(const float* __restrict__ sg,
                                               int* __restrict__ label) {
    int q = blockIdx.x * 256 + threadIdx.x;        // over B*H*W/4
    if (q >= BHW / 4) return;
    int b = q >> 12;                                // q / (HW/4)
    int hw4 = q & (HW / 4 - 1);
    int lab0 = 0, lab1 = 0, lab2 = 0, lab3 = 0;
    #pragma unroll
    for (int j = 0; j < Jj; ++j) {
        float4 v = ((const float4*)(sg + ((size_t)b * Jj + j) * HW))[hw4];
        if (v.x > 0.5f) lab0 = j;
        if (v.y > 0.5f) lab1 = j;
        if (v.z > 0.5f) lab2 = j;
        if (v.w > 0.5f) lab3 = j;
    }
    ((int4*)label)[q] = make_int4(lab0, lab1, lab2, lab3);
}

// ---------------- Kernel 3: pack 9 neighbor labels (3 bits each) ----------------
__global__ void __launch_bounds__(256) k_pack(const int* __restrict__ label,
                                              int* __restrict__ packed) {
    int idx = blockIdx.x * 256 + threadIdx.x;      // over B*H*W
    if (idx >= BHW) return;
    int b = idx >> 14, h = (idx >> 7) & 127, w = idx & 127;
    const int* lb = label + (size_t)b * HW;
    int code = 0;
    #pragma unroll
    for (int dy = -1; dy <= 1; ++dy)
        #pragma unroll
        for (int dx = -1; dx <= 1; ++dx) {
            int tap = (dy + 1) * 3 + (dx + 1);
            int hh = min(max(h + dy, 0), Hh - 1);  // clamped; OOB taps skipped later
            int ww = min(max(w + dx, 0), Ww - 1);
            code |= lb[hh * Ww + ww] << (3 * tap);
        }
    packed[idx] = code;
}

// ---------------- Kernel 4: style select + per-class FC + ReLU -> mu[b,j,s] ----------------
__global__ void __launch_bounds__(256) k_mu(const float* __restrict__ style_codes,
                                            const int* __restrict__ mask_codes,
                                            const float* __restrict__ W_mu,
                                            const float* __restrict__ b_mu,
                                            float* __restrict__ mu) {
    int bj = blockIdx.x;                 // 0..31
    int b = bj >> 3, j = bj & 7, o = threadIdx.x;
    __shared__ float code[Ss];
    int sel = (mask_codes[bj] == 1) ? j : Jj;        // j or shared code J
    code[o] = style_codes[((size_t)b * (Jj + 1) + sel) * Ss + o];
    __syncthreads();
    const float* wr = W_mu + ((size_t)j * Ss + o) * Ss;
    float acc = b_mu[j * Ss + o];
    for (int s = 0; s < Ss; ++s) acc = fmaf(code[s], wr[s], acc);
    mu[(size_t)bj * Ss + o] = fmaxf(acc, 0.f);
}

// ---------------- Kernel 5: WMMA tap GEMMs  V[(tensor,tap), n, c] ----------------
// V[t, tap, n=(b*8+j), c] = sum_s Wt[c, s, tap] * mu[n, s]
// One wave per 16x16 output tile; K-loop of 64 x v_wmma_f32_16x16x4_f32.
__global__ void __launch_bounds__(32) k_tap_gemm(const float* __restrict__ Wg,
                                                 const float* __restrict__ Wb,
                                                 const float* __restrict__ mu,
                                                 float* __restrict__ V) {
    int idx = blockIdx.x;
    int n0 = (idx & 1) * 16;  idx >>= 1;   // N tile: 0 or 16
    int c0 = (idx & 15) * 16; idx >>= 4;   // M tile: 16 of them
    int tap = idx % 9;
    int tensor = idx / 9;                  // 0 = gamma, 1 = beta
    const float* Wt = tensor ? Wb : Wg;

    int lane = threadIdx.x;
    int half = lane >> 4;                  // 0: lanes 0-15 (K, K+1), 1: lanes 16-31 (K+2, K+3)
    int lm = lane & 15;

    // A[m,k] = Wt[(c0+m)*2304 + k*9 + tap] ; B[k,n] = mu[(n0+n)*256 + k]
    const float* arow = Wt + (size_t)(c0 + lm) * WSTRIDE + tap;
    const float* brow = mu + (size_t)(n0 + lm) * Ss;

    v8f acc = {};
    for (int k0 = 0; k0 < Ss; k0 += 4) {
        int k = k0 + 2 * half;
        v2f a, bf;
        a.x = arow[(size_t)k * 9];
        a.y = arow[(size_t)(k + 1) * 9];
        bf.x = brow[k];
        bf.y = brow[k + 1];
        acc = __builtin_amdgcn_wmma_f32_16x16x4_f32(
            /*neg_a=*/false, a, /*neg_b=*/false, bf,
            /*c_mod=*/(short)0, acc, /*reuse_a=*/false, /*reuse_b=*/false);
    }
    // D layout: VGPR r -> (M = r + 8*half, N = lane&15)
    float* out = V + ((size_t)((tensor * 9 + tap) * NN + n0 + lm)) * Cc + c0 + 8 * half;
    #pragma unroll
    for (int r = 0; r < 8; ++r) out[r] = acc[r];
}

// ---------------- Kernel 6: fused BN apply + region modulation ----------------
// 4 pixels/thread via b128; gamma/beta table interleaved as float2 -> 9 ds_load_b64/pixel.
__global__ void __launch_bounds__(256) k_fuse(const float* __restrict__ fp,
                                              const float* __restrict__ V,
                                              const float* __restrict__ scale,
                                              const float* __restrict__ shift,
                                              const float* __restrict__ b_gamma,
                                              const float* __restrict__ b_beta,
                                              const int* __restrict__ packed,
                                              float* __restrict__ out) {
    int b = blockIdx.x >> 8;               // grid = B*C
    int c = blockIdx.x & (Cc - 1);
    int tid = threadIdx.x;

    __shared__ float sV[9][8][2];          // [tap][label][{gamma,beta}], 8B-aligned pairs
    if (tid < 144) {
        int tensor = tid / 72;
        int t = tid - tensor * 72;
        int tap = t >> 3, j = t & 7;
        sV[tap][j][tensor] =
            V[((size_t)((tensor * 9 + tap) * NN + b * Jj + j)) * Cc + c];
    }
    __syncthreads();

    float sc = scale[c], sh = shift[c], bg = b_gamma[c], bb = b_beta[c];
    const float4* fpc = (const float4*)(fp + ((size_t)b * Cc + c) * HW);
    float4*       oc  = (float4*)(out + ((size_t)b * Cc + c) * HW);
    const int4*   pk  = (const int4*)(packed + (size_t)b * HW);

    for (int q = tid; q < HW / 4; q += 256) {
        int p0 = q * 4;
        int h = p0 >> 7, w0 = p0 & 127;    // W=128 divisible by 4: all 4 pixels share row h
        float4 x = fpc[q];
        int4 cd = pk[q];
        float xs[4]  = {x.x, x.y, x.z, x.w};
        int   cds[4] = {cd.x, cd.y, cd.z, cd.w};
        float res[4];
        #pragma unroll
        for (int e = 0; e < 4; ++e) {
            int w = w0 + e;
            int code = cds[e];
            float gs = bg, bs = bb;
            #pragma unroll
            for (int dy = -1; dy <= 1; ++dy)
                #pragma unroll
                for (int dx = -1; dx <= 1; ++dx) {
                    int hh = h + dy, ww = w + dx;
                    if ((unsigned)hh < (unsigned)Hh && (unsigned)ww < (unsigned)Ww) {
                        int tap = (dy + 1) * 3 + (dx + 1);
                        int lab = (code >> (3 * tap)) & 7;
                        float2 v = *(const float2*)&sV[tap][lab][0];
                        gs += v.x;
                        bs += v.y;
                    }
                }
            float nf = fmaf(xs[e], sc, sh);
            res[e] = fmaf(nf, 1.f + gs, bs);
        }
        oc[q] = make_float4(res[0], res[1], res[2], res[3]);
    }
}

// ---------------- Launch ----------------
extern "C" void kernel_launch(void* const* d_in, const int* in_sizes, int n_in,
                              void* d_out, int out_size, void* d_ws, size_t ws_size,
                              hipStream_t stream) {
    const float* fp          = (const float*)d_in[0];
    const float* sg          = (const float*)d_in[1];
    const float* style_codes = (const float*)d_in[2];
    const int*   mask_codes  = (const int*)  d_in[3];
    const float* bn_gamma    = (const float*)d_in[4];
    const float* bn_beta     = (const float*)d_in[5];
    const float* W_mu        = (const float*)d_in[6];
    const float* b_mu        = (const float*)d_in[7];
    const float* W_gamma     = (const float*)d_in[8];
    const float* b_gamma     = (const float*)d_in[9];
    const float* W_beta      = (const float*)d_in[10];
    const float* b_beta      = (const float*)d_in[11];
    float* outp = (float*)d_out;

    // Workspace layout (floats/ints), total ~1.1 MB
    float* ws_scale = (float*)d_ws;                       // 256
    float* ws_shift = ws_scale + Cc;                      // 256
    float* ws_mu    = ws_shift + Cc;                      // 8192
    float* ws_V     = ws_mu + (size_t)NN * Ss;            // 2*9*32*256 = 147456
    int*   ws_label = (int*)(ws_V + (size_t)2 * 9 * NN * Cc); // 65536 ints
    int*   ws_pack  = ws_label + BHW;                     // 65536 ints

    k_bn_stats<<<Cc, 256, 0, stream>>>(fp, bn_gamma, bn_beta, ws_scale, ws_shift);
    k_label  <<<BHW / 4 / 256, 256, 0, stream>>>(sg, ws_label);
    k_pack   <<<BHW / 256, 256, 0, stream>>>(ws_label, ws_pack);
    k_mu     <<<NN, 256, 0, stream>>>(style_codes, mask_codes, W_mu, b_mu, ws_mu);
    k_tap_gemm<<<2 * 9 * 16 * 2, 32, 0, stream>>>(W_gamma, W_beta, ws_mu, ws_V);
    k_fuse   <<<Bn * Cc, 256, 0, stream>>>(fp, ws_V, ws_scale, ws_shift,
                                           b_gamma, b_beta, ws_pack, outp);
}